// TinyThetaGNN_47244640256095
// MI455X (gfx1250) — compile-verified
//
#include <hip/hip_runtime.h>
#include <hip/hip_bf16.h>

typedef __attribute__((ext_vector_type(2))) float v2f;
typedef __attribute__((ext_vector_type(8))) float v8f;

#define N_NODES 100000
#define N_EDGES 1200000
#define N_GRAPHS 256
#define HID 64

// ---------------- degree / norm ----------------
__global__ void deg_kernel(const int* __restrict__ dst, float* __restrict__ deg, int E) {
    int e = blockIdx.x * blockDim.x + threadIdx.x;
    if (e >= E) return;
    atomicAdd(&deg[dst[e]], 1.0f);
}

__global__ void dinv_kernel(float* __restrict__ deg, int n) {
    int v = blockIdx.x * blockDim.x + threadIdx.x;
    if (v >= n) return;
    // self-loop contributes +1, so degree is always >= 1
    deg[v] = rsqrtf(deg[v] + 1.0f);
}

// ---------------- layer-1 linear (K=5, trivial) ----------------
__global__ void lin1_kernel(const float* __restrict__ x, const float* __restrict__ W1,
                            float* __restrict__ out) {
    int t = blockIdx.x * blockDim.x + threadIdx.x;      // N*64
    if (t >= N_NODES * HID) return;
    int v = t >> 6, c = t & 63;
    const float* xr = x + v * 5;
    float s = 0.0f;
    #pragma unroll
    for (int k = 0; k < 5; ++k) s = fmaf(xr[k], W1[k * HID + c], s);
    out[t] = s;
}

// ---------------- edge aggregation (gather + atomic scatter) ----------------
__global__ __launch_bounds__(256) void agg_kernel(const float* __restrict__ h,
                                                  const int* __restrict__ src,
                                                  const int* __restrict__ dst,
                                                  const float* __restrict__ dinv,
                                                  float* __restrict__ out, int E) {
    int t = blockIdx.x * blockDim.x + threadIdx.x;
    int e = t >> 4;                 // 16 lanes per edge -> 64 channels via float4
    if (e >= E) return;
    int q = (t & 15) * 4;
    int s = src[e], d = dst[e];
    float nrm = dinv[s] * dinv[d];
    const float4 hv = *reinterpret_cast<const float4*>(h + (size_t)s * HID + q);
    float* o = out + (size_t)d * HID + q;
    atomicAdd(o + 0, hv.x * nrm);
    atomicAdd(o + 1, hv.y * nrm);
    atomicAdd(o + 2, hv.z * nrm);
    atomicAdd(o + 3, hv.w * nrm);
}

// ---------------- finalize: self-loop + bias + relu (in place on agg) ----------------
__global__ void fin_kernel(float* __restrict__ agg, const float* __restrict__ hlin,
                           const float* __restrict__ dinv, const float* __restrict__ bias) {
    int t = blockIdx.x * blockDim.x + threadIdx.x;      // N*64
    if (t >= N_NODES * HID) return;
    int v = t >> 6, c = t & 63;
    float dv = dinv[v];
    float val = agg[t] + dv * dv * hlin[t] + bias[c];
    agg[t] = fmaxf(val, 0.0f);
}

// ---------------- layer-2 linear: [N,64] @ [64,64] via V_WMMA_F32_16X16X4_F32 ------
// One wave computes one 16x16 output tile; K loop of 16 x (16x16x4) f32 WMMA ops.
// A fragment (16x4 f32, 2 VGPR): lane = hi*16+lo -> M=lo, K = 2*hi + vgpr
// B fragment (4x16 f32, 2 VGPR): lanes0-15 rows K={0,1}, lanes16-31 rows K={2,3}, N=lo
// C/D (16x16 f32, 8 VGPR): vgpr i -> M = i + 8*hi, N = lo
__global__ __launch_bounds__(256) void gemm_wmma_kernel(const float* __restrict__ H,
                                                        const float* __restrict__ W,
                                                        float* __restrict__ O) {
    int wave = blockIdx.x * (blockDim.x >> 5) + (threadIdx.x >> 5);
    int lane = threadIdx.x & 31;
    int nt = wave & 3;              // 4 N-tiles (64/16)
    int mt = wave >> 2;             // 6250 M-tiles (100000/16, exact)
    if (mt >= N_NODES / 16) return;
    int lo = lane & 15, hi = lane >> 4;
    int row = mt * 16 + lo;
    int col = nt * 16 + lo;

    v8f acc = {};
    const float* arow = H + (size_t)row * HID;
    #pragma unroll
    for (int k = 0; k < HID; k += 4) {
        v2f a, b;
        const float* ap = arow + k + 2 * hi;
        a.x = ap[0];
        a.y = ap[1];
        b.x = W[(k + 2 * hi) * HID + col];
        b.y = W[(k + 2 * hi + 1) * HID + col];
        acc = __builtin_amdgcn_wmma_f32_16x16x4_f32(
            /*neg_a=*/false, a, /*neg_b=*/false, b,
            /*c_mod=*/(short)0, acc, /*reuse_a=*/false, /*reuse_b=*/false);
    }
    #pragma unroll
    for (int i = 0; i < 8; ++i) {
        O[(size_t)(mt * 16 + i + 8 * hi) * HID + col] = acc[i];
    }
}

// ---------------- mean pool (atomics into [G,64] sums + counts) ----------------
__global__ void pool_kernel(const float* __restrict__ h, const int* __restrict__ batch,
                            float* __restrict__ sums, float* __restrict__ cnts) {
    int t = blockIdx.x * blockDim.x + threadIdx.x;      // N*64
    if (t >= N_NODES * HID) return;
    int v = t >> 6, c = t & 63;
    int g = batch[v];
    atomicAdd(&sums[g * HID + c], h[t]);
    if (c == 0) atomicAdd(&cnts[g], 1.0f);
}

// ---------------- final FC: [G,64] @ [64,2] + bias ----------------
__global__ void fc_kernel(const float* __restrict__ sums, const float* __restrict__ cnts,
                          const float* __restrict__ Wfc, const float* __restrict__ bfc,
                          float* __restrict__ out) {
    int t = blockIdx.x * blockDim.x + threadIdx.x;      // G*2
    if (t >= N_GRAPHS * 2) return;
    int g = t >> 1, o = t & 1;
    float inv = 1.0f / fmaxf(cnts[g], 1.0f);
    float s = bfc[o];
    #pragma unroll
    for (int c = 0; c < HID; ++c)
        s = fmaf(sums[g * HID + c] * inv, Wfc[c * 2 + o], s);
    out[t] = s;
}

extern "C" void kernel_launch(void* const* d_in, const int* in_sizes, int n_in,
                              void* d_out, int out_size, void* d_ws, size_t ws_size,
                              hipStream_t stream) {
    (void)in_sizes; (void)n_in; (void)out_size; (void)ws_size;
    const float* x    = (const float*)d_in[0];
    const int*   eidx = (const int*)d_in[1];     // [2, E] flat: src then dst
    const int*   batch= (const int*)d_in[2];
    const float* W1   = (const float*)d_in[3];
    const float* b1   = (const float*)d_in[4];
    const float* W2   = (const float*)d_in[5];
    const float* b2   = (const float*)d_in[6];
    const float* Wfc  = (const float*)d_in[7];
    const float* bfc  = (const float*)d_in[8];
    float* out = (float*)d_out;

    const int* src = eidx;
    const int* dst = eidx + N_EDGES;

    // workspace layout (floats)
    float* ws    = (float*)d_ws;
    float* dinv  = ws;                               // 100352 (deg then dinv, in place)
    float* bufA  = ws + 100352;                      // N*64: h1lin, then h2lin
    float* bufB  = bufA + (size_t)N_NODES * HID;     // N*64: agg1/h1, then agg2/h2
    float* sums  = bufB + (size_t)N_NODES * HID;     // G*64
    float* cnts  = sums + N_GRAPHS * HID;            // G

    const int B = 256;
    const int gE   = (N_EDGES + B - 1) / B;
    const int gN   = (N_NODES + B - 1) / B;
    const int gNH  = (N_NODES * HID + B - 1) / B;
    const int gAgg = (N_EDGES * 16 + B - 1) / B;
    const int gGemm = ((N_NODES / 16) * 4 + 7) / 8;  // 25000 waves, 8 waves/block

    // ---- normalization ----
    hipMemsetAsync(dinv, 0, (size_t)N_NODES * sizeof(float), stream);
    deg_kernel<<<gE, B, 0, stream>>>(dst, dinv, N_EDGES);
    dinv_kernel<<<gN, B, 0, stream>>>(dinv, N_NODES);

    // ---- layer 1 ----
    lin1_kernel<<<gNH, B, 0, stream>>>(x, W1, bufA);
    hipMemsetAsync(bufB, 0, (size_t)N_NODES * HID * sizeof(float), stream);
    agg_kernel<<<gAgg, B, 0, stream>>>(bufA, src, dst, dinv, bufB, N_EDGES);
    fin_kernel<<<gNH, B, 0, stream>>>(bufB, bufA, dinv, b1);     // bufB = h1

    // ---- layer 2 ----
    gemm_wmma_kernel<<<gGemm, B, 0, stream>>>(bufB, W2, bufA);   // bufA = h1 @ W2
    hipMemsetAsync(bufB, 0, (size_t)N_NODES * HID * sizeof(float), stream);
    agg_kernel<<<gAgg, B, 0, stream>>>(bufA, src, dst, dinv, bufB, N_EDGES);
    fin_kernel<<<gNH, B, 0, stream>>>(bufB, bufA, dinv, b2);     // bufB = h2

    // ---- pool + fc ----
    hipMemsetAsync(sums, 0, (size_t)(N_GRAPHS * HID + N_GRAPHS) * sizeof(float), stream);
    pool_kernel<<<gNH, B, 0, stream>>>(bufB, batch, sums, cnts);
    fc_kernel<<<(N_GRAPHS * 2 + B - 1) / B, B, 0, stream>>>(sums, cnts, Wfc, bfc, out);
}